// HierarchicalLLLoss_20667382629085
// MI455X (gfx1250) — compile-verified
//
#include <hip/hip_runtime.h>
#include <math.h>

// Hierarchical LL loss, MI455X (gfx1250, wave32).
// Main kernel: bandwidth-bound streaming block sums (131 MB @ 23.3 TB/s ~ 5.6us).
// Final kernel: partial reduction through V_WMMA_F32_16X16X4_F32 (A = ones =>
// D[m,n] = sum_k B[k,n] + C[m,n]; each WMMA folds 64 partials; operand
// placement inside B is irrelevant because only the grand total is used).

typedef float v2f __attribute__((ext_vector_type(2)));
typedef float v8f __attribute__((ext_vector_type(8)));

#define NUM_CLASSES 1000
#define VEC4_PER_ROW 250            // 1000 / 4
#define FULL_VEC4_ITERS 7           // 7*32 = 224 vec4s, unpredicated
#define TAIL_VEC4 26                // 250 - 224, lanes 0..25
#define IGNORE_INDEX (-100)
#define TOKENS_PER_WAVE 4
#define WAVES_PER_BLOCK 8
#define TOKENS_PER_BLOCK (TOKENS_PER_WAVE * WAVES_PER_BLOCK)

__device__ __forceinline__ float wave_sum32(float v) {
#pragma unroll
  for (int m = 16; m >= 1; m >>= 1) v += __shfl_xor(v, m, 32);
  return v;
}

__global__ __launch_bounds__(256)
void hll_partial_kernel(const float* __restrict__ inputs,
                        const int* __restrict__ target,
                        const float* __restrict__ weights,
                        float* __restrict__ partial_loss,
                        float* __restrict__ partial_cnt,
                        int ntok) {
  __shared__ float s_loss[WAVES_PER_BLOCK];
  __shared__ float s_cnt[WAVES_PER_BLOCK];

  const int lane = threadIdx.x & 31;
  const int wave = threadIdx.x >> 5;

  float wl = 0.f;   // per-wave loss accumulator (lane 0)
  float wc = 0.f;   // per-wave valid count (lane 0)

  const int tok0 = blockIdx.x * TOKENS_PER_BLOCK + wave * TOKENS_PER_WAVE;

  for (int i = 0; i < TOKENS_PER_WAVE; ++i) {
    const int tok = tok0 + i;
    if (tok >= ntok) break;                 // uniform within the wave
    const int t = target[tok];
    if (t == IGNORE_INDEX) continue;        // reference zeroes these tokens

    const float*  row  = inputs + (size_t)tok * NUM_CLASSES;
    const float4* row4 = (const float4*)row;   // rows are 4000 B => 16 B aligned

    // ---- s3: full row sum; 7 unconditional coalesced b128 loads + 1 tail ----
    float a3 = 0.f;
#pragma unroll
    for (int k = 0; k < FULL_VEC4_ITERS; ++k) {
      const float4 v = row4[k * 32 + lane];         // q <= 223 < 250 always
      a3 += (v.x + v.y) + (v.z + v.w);
    }
    if (lane < TAIL_VEC4) {
      const float4 v = row4[FULL_VEC4_ITERS * 32 + lane];
      a3 += (v.x + v.y) + (v.z + v.w);
    }

    // ---- s2: block of 100 containing t (b2 multiple of 100 => 25 aligned f4) ----
    const int b2 = (t / 100) * 100;
    float a2 = 0.f;
    if (lane < 25) {
      const float4 v = row4[b2 / 4 + lane];   // L0/L2 hit (just streamed)
      a2 = (v.x + v.y) + (v.z + v.w);
    }

    // ---- s1: block of 10 containing t ----
    const int b1 = (t / 10) * 10;
    float a1 = (lane < 10) ? row[b1 + lane] : 0.f;

    // ---- leaf probability ----
    const float xt = (lane == 0) ? row[t] : 0.f;

    const float s3 = wave_sum32(a3);
    const float s2 = wave_sum32(a2);
    const float s1 = wave_sum32(a1);

    if (lane == 0) {
      const float w0 = weights[t * 3 + 0];
      const float w1 = weights[t * 3 + 1];
      const float w2 = weights[t * 3 + 2];
      const float T0 = (xt != 0.f) ? -logf(xt / s1) : 0.f;
      const float T1 = (s1 != 0.f) ? -logf(s1 / s2) : 0.f;
      const float T2 = (s2 != 0.f) ? -logf(s2 / s3) : 0.f;
      wl += w0 * T0 + w1 * T1 + w2 * T2;
      wc += 1.f;
    }
  }

  if (lane == 0) { s_loss[wave] = wl; s_cnt[wave] = wc; }
  __syncthreads();
  if (threadIdx.x == 0) {
    float L = 0.f, C = 0.f;
#pragma unroll
    for (int w = 0; w < WAVES_PER_BLOCK; ++w) { L += s_loss[w]; C += s_cnt[w]; }
    partial_loss[blockIdx.x] = L;
    partial_cnt[blockIdx.x]  = C;
  }
}

// Final reduction: one wave. A = ones(16x4) so each V_WMMA_F32_16X16X4_F32
// computes D[m,n] = sum_k B[k,n] + C[m,n]; summing D row 0 over the 16 columns
// (c[0], lanes 0..15) yields the total of every value ever placed in B.
// Full groups of 64 use unpredicated global_load_b64 (contiguous v2f pairs);
// only a (normally absent) remainder uses predicated scalar loads.
__global__ __launch_bounds__(32)
void hll_final_kernel(const float* __restrict__ partial_loss,
                      const float* __restrict__ partial_cnt,
                      float* __restrict__ out,
                      int nblocks) {
  const int lane = threadIdx.x;

  v8f cl = {0.f, 0.f, 0.f, 0.f, 0.f, 0.f, 0.f, 0.f};
  v8f cc = {0.f, 0.f, 0.f, 0.f, 0.f, 0.f, 0.f, 0.f};
  v2f ones; ones.x = 1.f; ones.y = 1.f;

  const int full = nblocks >> 6;                 // groups of 64 values
  const v2f* pl2 = (const v2f*)partial_loss;     // 8B-aligned (padded offset)
  const v2f* pc2 = (const v2f*)partial_cnt;

  for (int it = 0; it < full; ++it) {
    const v2f bl = pl2[it * 32 + lane];          // one b64 load, no predication
    const v2f bc = pc2[it * 32 + lane];
    cl = __builtin_amdgcn_wmma_f32_16x16x4_f32(false, ones, false, bl,
                                               (short)0, cl, false, false);
    cc = __builtin_amdgcn_wmma_f32_16x16x4_f32(false, ones, false, bc,
                                               (short)0, cc, false, false);
  }

  const int base = full << 6;
  if (base < nblocks) {                          // tail (not taken for 32768 toks)
    const int q0 = base + lane;
    const int q1 = q0 + 32;
    v2f bl, bc;
    bl.x = (q0 < nblocks) ? partial_loss[q0] : 0.f;
    bl.y = (q1 < nblocks) ? partial_loss[q1] : 0.f;
    bc.x = (q0 < nblocks) ? partial_cnt[q0] : 0.f;
    bc.y = (q1 < nblocks) ? partial_cnt[q1] : 0.f;
    cl = __builtin_amdgcn_wmma_f32_16x16x4_f32(false, ones, false, bl,
                                               (short)0, cl, false, false);
    cc = __builtin_amdgcn_wmma_f32_16x16x4_f32(false, ones, false, bc,
                                               (short)0, cc, false, false);
  }

  float L = cl[0];   // row M=0: lanes 0..15 hold the 16 column sums
  float C = cc[0];
#pragma unroll
  for (int m = 8; m >= 1; m >>= 1) {
    L += __shfl_xor(L, m, 32);
    C += __shfl_xor(C, m, 32);
  }
  if (lane == 0) out[0] = L / fmaxf(C, 1.0f);
}

extern "C" void kernel_launch(void* const* d_in, const int* in_sizes, int n_in,
                              void* d_out, int out_size, void* d_ws, size_t ws_size,
                              hipStream_t stream) {
  // setup_inputs() order: inputs, target, onehot_num, onehot_den, weights.
  const float* inputs  = (const float*)d_in[0];
  const int*   target  = (const int*)d_in[1];
  const float* weights = (const float*)d_in[4];   // one-hot tensors never needed

  const int ntok    = in_sizes[1];
  const int nblocks = (ntok + TOKENS_PER_BLOCK - 1) / TOKENS_PER_BLOCK;  // 1024
  const int cnt_off = (nblocks + 1) & ~1;         // keep 8B alignment for b64 loads

  float* partial_loss = (float*)d_ws;             // nblocks floats
  float* partial_cnt  = partial_loss + cnt_off;   // nblocks floats (~8 KB total)

  hll_partial_kernel<<<nblocks, 256, 0, stream>>>(inputs, target, weights,
                                                  partial_loss, partial_cnt, ntok);
  hll_final_kernel<<<1, 32, 0, stream>>>(partial_loss, partial_cnt,
                                         (float*)d_out, nblocks);
}